// RoIBlock_11192684773426
// MI455X (gfx1250) — compile-verified
//
#include <hip/hip_runtime.h>
#include <stdint.h>

// roi_pool (torchvision semantics) for MI455X / gfx1250.
// B=4, C=256, H=W=50, R=256, out=7x7, scale=1/16. fp32 in/out.
//
// One block per (roi, ph). Stage the bin-row's feature rows
// (<=9 rows x 50 cols x 32 channels = 57.6KB) into LDS with the CDNA5
// async global->LDS DMA path (global_load_async_to_lds_b64 / ASYNCcnt /
// s_wait_asynccnt), then 32x7 threads reduce their (channel, pw) bin
// from LDS. Load mapping uses only power-of-two decompositions so no
// runtime integer division appears in the hot loop.

#define ROI_OUT 7
#define C_CH    256
#define H_F     50
#define W_F     50
#define SCALE   0.0625f
#define CHUNK   32        // channels staged per LDS tile
#define MAXROWS 9         // ceil(51/7)+1 upper bound on rows per bin
#define TPC     8         // threads per channel in the load phase (256/CHUNK)

__device__ __forceinline__ void async_ld_b64(uint32_t lds_off, uint32_t goff,
                                             const float* base) {
  // GVS mode: global address is the 64-bit SGPR base plus the 32-bit offset goff; LDS address is lds_off.
  asm volatile("global_load_async_to_lds_b64 %0, %1, %2"
               :: "v"(lds_off), "v"(goff), "s"(base)
               : "memory");
}

__device__ __forceinline__ void wait_async0() {
  asm volatile("s_wait_asynccnt 0" ::: "memory");
}

__global__ __launch_bounds__(256) void roi_pool_gfx1250(
    const float* __restrict__ x,      // [B, C, H, W]
    const float* __restrict__ rois,   // [R, 4] (y1, x1, y2, x2) image coords
    const int*   __restrict__ roi_idx,// [R]
    float*       __restrict__ out,    // [R, C, 7, 7]
    int R) {
  __shared__ __align__(16) float tile[CHUNK * MAXROWS * W_F];  // 57.6 KB

  const int blk = blockIdx.x;
  const int r   = blk / ROI_OUT;
  const int ph  = blk % ROI_OUT;
  if (r >= R) return;

  const int tid = threadIdx.x;
  const int n   = roi_idx[r];

  // Reference: _round(v) = floor(v + 0.5) on v = roi * scale
  const float y1 = floorf(rois[r * 4 + 0] * SCALE + 0.5f);
  const float x1 = floorf(rois[r * 4 + 1] * SCALE + 0.5f);
  const float y2 = floorf(rois[r * 4 + 2] * SCALE + 0.5f);
  const float x2 = floorf(rois[r * 4 + 3] * SCALE + 0.5f);
  const float roi_h = fmaxf(y2 - y1 + 1.0f, 1.0f);
  const float roi_w = fmaxf(x2 - x1 + 1.0f, 1.0f);
  const float bh = roi_h / (float)ROI_OUT;
  const float bw = roi_w / (float)ROI_OUT;

  // Row range for this ph (uniform across block), clipped to [0, H]
  const float hs_f = fminf(fmaxf(floorf((float)ph * bh) + y1, 0.0f), (float)H_F);
  const float he_f = fminf(fmaxf(ceilf((float)(ph + 1) * bh) + y1, 0.0f), (float)H_F);
  const int hs    = (int)hs_f;
  const int he    = (int)he_f;
  const int nrows = he - hs;          // 0..9 (uniform per block)

  // ---- load-phase thread mapping (all power-of-two, no division) ----
  const int cl   = tid >> 3;          // channel-in-chunk, 0..31
  const int lane = tid & (TPC - 1);   // 0..7
  const int pairs_per_ch = nrows * (W_F / 2);             // b64 pairs per channel
  const uint32_t tile_base = (uint32_t)(size_t)tile;      // low 32 bits = LDS offset
  const uint32_t lds_ch_base = tile_base + (uint32_t)(cl * nrows * W_F * 4);

  // ---- compute-phase thread mapping ----
  const int c_loc = tid / ROI_OUT;    // constant divisor -> cheap
  const int pw    = tid % ROI_OUT;
  const bool active = (tid < CHUNK * ROI_OUT);
  const float ws_f = fminf(fmaxf(floorf((float)pw * bw) + x1, 0.0f), (float)W_F);
  const float we_f = fminf(fmaxf(ceilf((float)(pw + 1) * bw) + x1, 0.0f), (float)W_F);
  const int ws = (int)ws_f;
  const int we = (int)we_f;

  for (int c0 = 0; c0 < C_CH; c0 += CHUNK) {
    __syncthreads();  // protect LDS tile from previous iteration's readers

    if (nrows > 0) {
      // Rows hs..he-1 are contiguous per channel (full-width rows), so each
      // channel is one contiguous run of nrows*50 floats in global memory.
      const uint32_t gch =
          (uint32_t)(((((n * C_CH + c0 + cl) * H_F) + hs) * W_F) * 4);
      for (int p = lane; p < pairs_per_ch; p += TPC) {
        const uint32_t boff = (uint32_t)(p << 3);  // 8B per b64 pair
        async_ld_b64(lds_ch_base + boff, gch + boff, x);
      }
      wait_async0();  // each wave drains its own ASYNCcnt before the barrier
    }
    __syncthreads();

    if (active) {
      float m;
      if (nrows <= 0 || we <= ws) {
        m = 0.0f;  // empty bin -> 0 per reference
      } else {
        m = -3.402823466e38f;
        const float* tch = &tile[c_loc * nrows * W_F];
        for (int row = 0; row < nrows; ++row) {
          const float* trow = tch + row * W_F;
          for (int w = ws; w < we; ++w) m = fmaxf(m, trow[w]);
        }
      }
      const int c = c0 + c_loc;
      out[((r * C_CH + c) * ROI_OUT + ph) * ROI_OUT + pw] = m;
    }
  }
}

extern "C" void kernel_launch(void* const* d_in, const int* in_sizes, int n_in,
                              void* d_out, int out_size, void* d_ws, size_t ws_size,
                              hipStream_t stream) {
  const float* x       = (const float*)d_in[0];  // [4,256,50,50] fp32
  const float* rois    = (const float*)d_in[1];  // [R,4] fp32
  const int*   roi_idx = (const int*)d_in[2];    // [R] int32
  float*       out     = (float*)d_out;          // [R,256,7,7] fp32

  const int R = in_sizes[2];
  dim3 grid(R * ROI_OUT);
  dim3 block(256);
  roi_pool_gfx1250<<<grid, block, 0, stream>>>(x, rois, roi_idx, out, R);
}